// MoEGate_5746666242505
// MI455X (gfx1250) — compile-verified
//
#include <hip/hip_runtime.h>
#include <stdint.h>

typedef float v2f __attribute__((ext_vector_type(2)));
typedef float v8f __attribute__((ext_vector_type(8)));
typedef int   v4i __attribute__((ext_vector_type(4)));

typedef __attribute__((address_space(1))) v4i* gptr_v4i;
typedef __attribute__((address_space(3))) v4i* lptr_v4i;

namespace {
constexpr int kH      = 7168;   // hidden dim
constexpr int kE      = 256;    // experts
constexpr int kTopK   = 8;
constexpr int kKC     = 64;     // K chunk
constexpr int kAPitch = 68;     // LDS pitch for A tile (conflict-free ds_load_b64)
constexpr int kSPitch = 260;    // LDS pitch for score tile (conflict-free scatter)
constexpr int kMT     = 32;     // tokens per block
}

// ---- CDNA5 async global->LDS copy (B128) ----------------------------------
__device__ __forceinline__ void async_copy_b128(const float* gptr, float* lptr) {
#if __has_builtin(__builtin_amdgcn_global_load_async_to_lds_b128)
    __builtin_amdgcn_global_load_async_to_lds_b128(
        (gptr_v4i)(uintptr_t)gptr,
        (lptr_v4i)(unsigned)(uintptr_t)lptr,
        0, 0);
#else
    unsigned lds_off = (unsigned)(uintptr_t)lptr;   // generic LDS ptr low 32b == AS3 offset
    asm volatile("global_load_async_to_lds_b128 %0, %1, off"
                 :: "v"(lds_off), "v"(gptr) : "memory");
#endif
}

__device__ __forceinline__ void wait_async0() {
#if __has_builtin(__builtin_amdgcn_s_wait_asynccnt)
    __builtin_amdgcn_s_wait_asynccnt(0);
#else
    asm volatile("s_wait_asynccnt 0x0" ::: "memory");
#endif
}

__global__ __launch_bounds__(512)
void moe_gate_fused(const float* __restrict__ hidden,
                    const float* __restrict__ weight,
                    const float* __restrict__ bias,
                    int*   __restrict__ out_idx,
                    float* __restrict__ out_w)
{
    __shared__ float As[2][kMT * kAPitch];   // double-buffered A tile (32 x 64)
    __shared__ float Sc[kMT * kSPitch];      // sigmoid scores (32 x 256)

    const int tid  = threadIdx.x;
    const int lane = tid & 31;
    const int wave = tid >> 5;               // 0..15
    const int tok_base = blockIdx.x * kMT;
    const int eb   = wave * 16;              // expert tile base for this wave

    const int lrow = lane & 15;
    const int koff = (lane < 16) ? 0 : 2;    // K sub-offset per WMMA fragment layout

    // cooperative A loader: 32 rows x 64 cols = 2048 floats / 512 threads = 1 b128 each
    const int arow = tid >> 4;               // 0..31
    const int acol = (tid & 15) * 4;         // 0..60
    const float* hrow = hidden + (size_t)(tok_base + arow) * kH + acol;
    float* adst0 = &As[0][arow * kAPitch + acol];
    float* adst1 = &As[1][arow * kAPitch + acol];

    v8f acc0 = {};  // tokens 0..15  x experts eb..eb+15
    v8f acc1 = {};  // tokens 16..31 x experts eb..eb+15

    const float* wrow = weight + (size_t)(eb + lrow) * kH + koff;

    async_copy_b128(hrow, adst0);            // prefetch chunk 0 into buffer 0
    constexpr int nch = kH / kKC;            // 112 chunks
    for (int c = 0; c < nch; ++c) {
        wait_async0();                       // my async LDS writes have landed
        __syncthreads();                     // everyone's writes visible / reads done
        if (c + 1 < nch)
            async_copy_b128(hrow + (c + 1) * kKC, ((c + 1) & 1) ? adst1 : adst0);
        const float* ab    = &As[c & 1][0];
        const float* bbase = wrow + c * kKC;
#pragma unroll
        for (int kk = 0; kk < kKC; kk += 4) {
            v2f b;
            b.x = bbase[kk];
            b.y = bbase[kk + 1];
            const float* ap0 = ab + lrow * kAPitch + kk + koff;
            v2f a0; a0.x = ap0[0]; a0.y = ap0[1];
            const float* ap1 = ap0 + 16 * kAPitch;
            v2f a1; a1.x = ap1[0]; a1.y = ap1[1];
            acc0 = __builtin_amdgcn_wmma_f32_16x16x4_f32(
                false, a0, false, b, (short)0, acc0, false, false);
            acc1 = __builtin_amdgcn_wmma_f32_16x16x4_f32(
                false, a1, false, b, (short)0, acc1, false, false);
        }
    }

    // sigmoid + scatter to LDS score tile (C-matrix layout: VGPR r -> M=r / r+8)
#pragma unroll
    for (int r = 0; r < 8; ++r) {
        const int t0 = (lane < 16) ? r : (r + 8);
        const int e  = eb + lrow;
        Sc[t0 * kSPitch + e]        = 1.f / (1.f + __expf(-acc0[r]));
        Sc[(t0 + 16) * kSPitch + e] = 1.f / (1.f + __expf(-acc1[r]));
    }
    __syncthreads();

    // ---- gating: wave w handles tokens w and w+16; lane owns experts lane*8..lane*8+7 ----
    float bia[8];
#pragma unroll
    for (int j = 0; j < 8; ++j) bia[j] = bias[lane * 8 + j];

    const float NEG_INF = -__builtin_inff();

    for (int half = 0; half < 2; ++half) {
        const int t = wave + half * 16;
        const float* sr = &Sc[t * kSPitch + lane * 8];
        float s[8], scb[8];
#pragma unroll
        for (int j = 0; j < 8; ++j) { s[j] = sr[j]; scb[j] = s[j] + bia[j]; }

        // lane-local top-2 of scores_for_choice
        float m1 = NEG_INF, m2 = NEG_INF;
#pragma unroll
        for (int j = 0; j < 8; ++j) {
            const float v = scb[j];
            if (v > m1) { m2 = m1; m1 = v; }
            else if (v > m2) { m2 = v; }
        }
        // merge top-2 across the 4 lanes of each group (experts/group = 32 = 4 lanes)
#pragma unroll
        for (int off = 1; off <= 2; off <<= 1) {
            const float o1 = __shfl_xor(m1, off, 32);
            const float o2 = __shfl_xor(m2, off, 32);
            const float n1 = fmaxf(m1, o1);
            const float n2 = fmaxf(fminf(m1, o1), fmaxf(m2, o2));
            m1 = n1; m2 = n2;
        }
        const float gscore = m1 + m2;
        float gv[8];
#pragma unroll
        for (int g = 0; g < 8; ++g) gv[g] = __shfl(gscore, g * 4, 32);

        // top-4 groups (strict > keeps lowest index on ties, like lax.top_k)
        int gmask = 0;
#pragma unroll
        for (int rsel = 0; rsel < 4; ++rsel) {
            float best = NEG_INF; int bi = 0;
#pragma unroll
            for (int g = 0; g < 8; ++g) {
                const bool ok = !((gmask >> g) & 1);
                if (ok && gv[g] > best) { best = gv[g]; bi = g; }
            }
            gmask |= (1 << bi);
        }

        const bool active = (gmask >> (lane >> 2)) & 1;
        unsigned rem = active ? 0xFFu : 0u;

        float myw = 0.f; int myidx = 0; float wsum = 0.f;
        for (int r = 0; r < kTopK; ++r) {
            float bv = NEG_INF; int bidx = 0x7FFFFFFF; float bw = 0.f;
#pragma unroll
            for (int j = 0; j < 8; ++j) {
                if ((rem >> j) & 1) {
                    if (scb[j] > bv) { bv = scb[j]; bidx = lane * 8 + j; bw = s[j]; }
                }
            }
            // wave-wide argmax: value desc, expert-index asc on ties
#pragma unroll
            for (int off = 16; off >= 1; off >>= 1) {
                const float ov = __shfl_xor(bv, off, 32);
                const int   oi = __shfl_xor(bidx, off, 32);
                const float ow = __shfl_xor(bw, off, 32);
                if (ov > bv || (ov == bv && oi < bidx)) { bv = ov; bidx = oi; bw = ow; }
            }
            if ((bidx >> 3) == lane) rem &= ~(1u << (bidx & 7));
            if (lane == r) { myidx = bidx; myw = bw; }
            wsum += bw;
        }

        const int tg = tok_base + t;
        if (lane < kTopK) {
            out_idx[tg * kTopK + lane] = myidx;
            out_w[tg * kTopK + lane]   = (myw / (wsum + 1e-20f)) * 2.5f;
        }
    }
}

extern "C" void kernel_launch(void* const* d_in, const int* in_sizes, int n_in,
                              void* d_out, int out_size, void* d_ws, size_t ws_size,
                              hipStream_t stream) {
    const float* hidden = (const float*)d_in[0];
    const float* weight = (const float*)d_in[1];
    const float* bias   = (const float*)d_in[2];

    const int E = in_sizes[2];               // 256
    const int H = in_sizes[1] / E;           // 7168
    const int T = in_sizes[0] / H;           // 16384

    int*   out_idx = (int*)d_out;
    float* out_w   = (float*)d_out + (size_t)T * kTopK;

    dim3 grid(T / kMT);
    moe_gate_fused<<<grid, 512, 0, stream>>>(hidden, weight, bias, out_idx, out_w);
}